// LoopEncoder_30485677867836
// MI455X (gfx1250) — compile-verified
//
#include <hip/hip_runtime.h>
#include <hip/hip_bf16.h>
#include <math.h>
#include <stdint.h>

typedef float v2f __attribute__((ext_vector_type(2)));
typedef float v8f __attribute__((ext_vector_type(8)));

#define NCH     64      // channels per conv layer (C+2 = 64 for conv0)
#define KDIM    192     // 3 taps * 64 channels (tap-major K ordering)
#define WSTRIDE 196     // LDS weight row stride (dwords): lanes 4 banks apart -> conflict-free b64
#define XCOLS   33      // input window for 16 outputs: 2*16+1
#define XSTRIDE 66      // LDS window stride (dwords): even (b64 align) + conflict-free
#define EDGES   32768
#define NPTS    32768
#define HIMG    256
#define CIMG    62

// ---------------------------------------------------------------------------
// Kernel 1: edge lengths + inclusive prefix sum (el2) + total length.
// ---------------------------------------------------------------------------
__global__ void edge_scan_kernel(const float* __restrict__ corners,
                                 float* __restrict__ el2,
                                 float* __restrict__ lsum) {
    __shared__ float s[1024];
    const int t  = threadIdx.x;
    const int CH = EDGES / 1024;          // 32
    const int base = t * CH;

    float acc = 0.f;
    for (int i = 0; i < CH; ++i) {
        int e  = base + i;
        int e2 = (e + 1 == EDGES) ? 0 : e + 1;
        float dx = corners[2*e2]     - corners[2*e];
        float dy = corners[2*e2 + 1] - corners[2*e + 1];
        acc += sqrtf(dx*dx + dy*dy);
    }
    s[t] = acc;
    __syncthreads();

    for (int off = 1; off < 1024; off <<= 1) {
        float v = (t >= off) ? s[t - off] : 0.f;
        __syncthreads();
        s[t] += v;
        __syncthreads();
    }
    float run = (t == 0) ? 0.f : s[t - 1];

    for (int i = 0; i < CH; ++i) {
        int e  = base + i;
        int e2 = (e + 1 == EDGES) ? 0 : e + 1;
        float dx = corners[2*e2]     - corners[2*e];
        float dy = corners[2*e2 + 1] - corners[2*e + 1];
        run += sqrtf(dx*dx + dy*dy);
        el2[e] = run;
    }
    if (t == 1023) lsum[0] = s[1023];
}

// ---------------------------------------------------------------------------
// Kernel 2: binary-search arc-length resampling + 62-channel gather.
// ---------------------------------------------------------------------------
__global__ void points_gather_kernel(const float* __restrict__ corners,
                                     const float* __restrict__ image,
                                     const float* __restrict__ el2,
                                     const float* __restrict__ lsum,
                                     float* __restrict__ xbuf) {
    const int p = blockIdx.x * blockDim.x + threadIdx.x;
    if (p >= NPTS) return;

    const float offset = (float)p * (lsum[0] * (1.0f / (float)NPTS));

    int lo = 0, hi = EDGES;
    while (lo < hi) {
        int mid = (lo + hi) >> 1;
        if (el2[mid] > offset) hi = mid; else lo = mid + 1;
    }
    int e = min(lo, EDGES - 1);

    float el1 = (e == 0) ? 0.f : el2[e - 1];
    int   e2  = (e + 1 == EDGES) ? 0 : e + 1;
    float sx = corners[2*e],  sy = corners[2*e + 1];
    float dx = corners[2*e2] - sx, dy = corners[2*e2 + 1] - sy;
    float len = sqrtf(dx*dx + dy*dy);
    float tt  = (offset - el1) / fmaxf(len, 0.0001f);
    float px = sx + tt * dx;
    float py = sy + tt * dy;

    int pr0 = min(max((int)rintf(px * (float)HIMG), 0), HIMG - 1);
    int pr1 = min(max((int)rintf(py * (float)HIMG), 0), HIMG - 1);

    const float* base = image + pr0 * HIMG + pr1;
    #pragma unroll 2
    for (int c = 0; c < CIMG; ++c)
        xbuf[c * NPTS + p] = base[c * HIMG * HIMG];
    xbuf[62 * NPTS + p] = px;
    xbuf[63 * NPTS + p] = py;
}

// ---------------------------------------------------------------------------
// Kernel 3/4/5: conv1d (reflect pad 1, stride 2, k=3) + bias + ReLU as WMMA:
//   y(64 x Lout) = W(64 x 192) * im2col(x)(192 x Lout), K tap-major.
// 128 threads = 4 wave32; wave owns one 64(M) x 16(N) tile.
//  - x window (64ch x 33col) pulled into LDS with global_load_async_to_lds_b32
//    (ASYNCcnt), overlapped with cooperative weight staging.
//  - 48 K-steps fully unrolled -> 192 v_wmma_f32_16x16x4_f32 per wave.
// ---------------------------------------------------------------------------
__global__ void conv_wmma_kernel(const float* __restrict__ x,
                                 const float* __restrict__ W,
                                 const float* __restrict__ bias,
                                 float* __restrict__ y,
                                 int Lin, int Lout) {
    __shared__ float Wl[NCH * WSTRIDE];       // 49 KB: weights, tap-major K, padded rows
    __shared__ float xs[4][XCOLS * XSTRIDE];  // 34 KB: per-wave x window, xs[u*66 + ch]

    const int tid   = threadIdx.x;            // 0..127
    const int lane  = tid & 31;
    const int wave  = tid >> 5;
    const int n     = lane & 15;              // column within N-tile
    const int hicol = lane >> 4;              // 0: lanes 0-15, 1: lanes 16-31

    const int numTiles = Lout >> 4;
    const int tile = blockIdx.x * 4 + wave;
    const bool active = tile < numTiles;
    const int j0 = (active ? tile : 0) << 4;

    // ---- Phase 1: async global->LDS fill of this wave's window (ASYNCcnt) ----
    if (active) {
        // columns u = lane (0..31): input index 2*j0-1+u; only -1 can occur -> reflect to +1
        int col = 2 * j0 - 1 + lane;
        col = col < 0 ? -col : col;
        const float* gp = x + col;
        float* lp = &xs[wave][lane * XSTRIDE];
        #pragma unroll 8
        for (int i = 0; i < NCH; ++i) {
            unsigned     dst = (unsigned)(uintptr_t)(lp + i);
            const float* src = gp + (size_t)i * Lin;
            asm volatile("global_load_async_to_lds_b32 %0, %1, off"
                         :: "v"(dst), "v"(src) : "memory");
        }
        // tail column u = 32 (index 2*j0+31 <= Lin-1, never reflects): lane = channel
        const float* gt = x + (2 * j0 + 31);
        unsigned     d0 = (unsigned)(uintptr_t)(&xs[wave][32 * XSTRIDE + lane]);
        const float* s0 = gt + (size_t)lane * Lin;
        asm volatile("global_load_async_to_lds_b32 %0, %1, off" :: "v"(d0), "v"(s0) : "memory");
        unsigned     d1 = (unsigned)(uintptr_t)(&xs[wave][32 * XSTRIDE + lane + 32]);
        const float* s1 = gt + (size_t)(lane + 32) * Lin;
        asm volatile("global_load_async_to_lds_b32 %0, %1, off" :: "v"(d1), "v"(s1) : "memory");
    }

    // ---- Phase 2 (overlaps the async fill): stage weights tap-major ----
    // Wl[m][k*64 + i] = W[m][i][k]
    for (int f = tid; f < NCH * KDIM; f += 128) {
        int m = f / KDIM;
        int r = f - m * KDIM;                 // K' = k*64 + i
        int k = r >> 6;
        int i = r & 63;
        Wl[m * WSTRIDE + r] = W[m * KDIM + i * 3 + k];
    }
    __syncthreads();

    if (!active) return;

    asm volatile("s_wait_asynccnt 0x0" ::: "memory");

    const int j = j0 + n;
    const int mbase = (lane & 15) * WSTRIDE;
    v8f c0 = {}, c1 = {}, c2 = {}, c3 = {};

    #pragma unroll
    for (int kk = 0; kk < KDIM / 4; ++kk) {
        const int Kb = kk * 4 + 2 * hicol;    // this lane covers K' = Kb, Kb+1
        const int k  = Kb >> 6;               // tap
        const int i  = Kb & 63;               // channel (even -> b64 aligned)

        // B fragment: adjacent channels, same tap -> one ds_load_b64
        v2f b  = *reinterpret_cast<const v2f*>(&xs[wave][(2 * n + k) * XSTRIDE + i]);
        // A fragments: K' pair contiguous in padded weight rows -> ds_load_b64
        v2f a0 = *reinterpret_cast<const v2f*>(&Wl[mbase + Kb]);
        v2f a1 = *reinterpret_cast<const v2f*>(&Wl[mbase + 16 * WSTRIDE + Kb]);
        v2f a2 = *reinterpret_cast<const v2f*>(&Wl[mbase + 32 * WSTRIDE + Kb]);
        v2f a3 = *reinterpret_cast<const v2f*>(&Wl[mbase + 48 * WSTRIDE + Kb]);

        c0 = __builtin_amdgcn_wmma_f32_16x16x4_f32(false, a0, false, b, (short)0, c0, false, false);
        c1 = __builtin_amdgcn_wmma_f32_16x16x4_f32(false, a1, false, b, (short)0, c1, false, false);
        c2 = __builtin_amdgcn_wmma_f32_16x16x4_f32(false, a2, false, b, (short)0, c2, false, false);
        c3 = __builtin_amdgcn_wmma_f32_16x16x4_f32(false, a3, false, b, (short)0, c3, false, false);
    }

    // C/D layout: VGPR r holds M=r (lanes 0-15) / M=r+8 (lanes 16-31)
    #pragma unroll
    for (int r = 0; r < 8; ++r) {
        int m0 = r + 8 * hicol;
        y[(m0     ) * Lout + j] = fmaxf(c0[r] + bias[m0     ], 0.f);
        y[(m0 + 16) * Lout + j] = fmaxf(c1[r] + bias[m0 + 16], 0.f);
        y[(m0 + 32) * Lout + j] = fmaxf(c2[r] + bias[m0 + 32], 0.f);
        y[(m0 + 48) * Lout + j] = fmaxf(c3[r] + bias[m0 + 48], 0.f);
    }
}

// ---------------------------------------------------------------------------
// Kernel 6: global max over length axis + FC(relu) + FC(sigmoid).
// ---------------------------------------------------------------------------
__global__ void head_kernel(const float* __restrict__ y2,   // 64 x 4096
                            const float* __restrict__ Wf,
                            const float* __restrict__ bf,
                            const float* __restrict__ Wp,
                            const float* __restrict__ bp,
                            float* __restrict__ out) {
    __shared__ float red[256];
    __shared__ float mx[64];
    __shared__ float feat[64];

    const int t = threadIdx.x;
    const int m = t >> 2;
    const int q = t & 3;

    const float* row = y2 + m * 4096 + q * 1024;
    float v = -INFINITY;
    for (int i = 0; i < 1024; ++i) v = fmaxf(v, row[i]);
    red[t] = v;
    __syncthreads();

    if (q == 0)
        mx[m] = fmaxf(fmaxf(red[t], red[t + 1]), fmaxf(red[t + 2], red[t + 3]));
    __syncthreads();

    if (t < 64) {
        float acc = bf[t];
        for (int k = 0; k < 64; ++k) acc += mx[k] * Wf[t * 64 + k];
        acc = fmaxf(acc, 0.f);
        feat[t] = acc;
        out[t]  = acc;
    }
    __syncthreads();

    if (t == 0) {
        float acc = bp[0];
        for (int k = 0; k < 64; ++k) acc += feat[k] * Wp[k];
        out[64] = 1.f / (1.f + __expf(-acc));
    }
}

// ---------------------------------------------------------------------------
extern "C" void kernel_launch(void* const* d_in, const int* in_sizes, int n_in,
                              void* d_out, int out_size, void* d_ws, size_t ws_size,
                              hipStream_t stream) {
    const float* image   = (const float*)d_in[0];
    const float* corners = (const float*)d_in[1];
    const float* W0 = (const float*)d_in[2];
    const float* b0 = (const float*)d_in[3];
    const float* W1 = (const float*)d_in[4];
    const float* b1 = (const float*)d_in[5];
    const float* W2 = (const float*)d_in[6];
    const float* b2 = (const float*)d_in[7];
    const float* Wf = (const float*)d_in[8];
    const float* bf = (const float*)d_in[9];
    const float* Wp = (const float*)d_in[10];
    const float* bp = (const float*)d_in[11];
    float* out = (float*)d_out;

    float* ws   = (float*)d_ws;
    float* el2  = ws;                              // EDGES
    float* lsum = ws + EDGES;                      // 1 (+pad)
    float* xbuf = ws + EDGES + 16;                 // 64 * 32768
    float* y0   = xbuf + NCH * NPTS;               // 64 * 16384
    float* y1   = y0   + NCH * (NPTS / 2);         // 64 * 8192
    float* y2   = y1   + NCH * (NPTS / 4);         // 64 * 4096

    edge_scan_kernel<<<1, 1024, 0, stream>>>(corners, el2, lsum);

    points_gather_kernel<<<NPTS / 256, 256, 0, stream>>>(corners, image, el2, lsum, xbuf);

    conv_wmma_kernel<<<(NPTS / 2) / 16 / 4, 128, 0, stream>>>(xbuf, W0, b0, y0, NPTS,     NPTS / 2);
    conv_wmma_kernel<<<(NPTS / 4) / 16 / 4, 128, 0, stream>>>(y0,   W1, b1, y1, NPTS / 2, NPTS / 4);
    conv_wmma_kernel<<<(NPTS / 8) / 16 / 4, 128, 0, stream>>>(y1,   W2, b2, y2, NPTS / 4, NPTS / 8);

    head_kernel<<<1, 256, 0, stream>>>(y2, Wf, bf, Wp, bp, out);
}